// CostModelV2_27728308863587
// MI455X (gfx1250) — compile-verified
//
#include <hip/hip_runtime.h>
#include <hip/hip_bf16.h>

#define NN 50000   // multiple of 16 -> full tiles only (guarded per-wave)
#define EE 1250000
#define FF 176
#define HH 64
#define H2 128
#define H2P 132    // padded LDS stride for the 16x128 staging tile

typedef __attribute__((ext_vector_type(2))) float v2f;
typedef __attribute__((ext_vector_type(8))) float v8f;

// D = A(16x4, f32) * B(4x16, f32) + C   (true fp32 WMMA on CDNA5)
__device__ __forceinline__ v8f wmma4(v2f a, v2f b, v8f c) {
  return __builtin_amdgcn_wmma_f32_16x16x4_f32(
      false, a, false, b, (short)0, c, false, false);
}

__device__ __forceinline__ void atom_fadd(float* p, float v) {
  unsafeAtomicAdd(p, v);  // native global_atomic_add_f32 (no CAS loop)
}

// ---------------------------------------------------------------------------
// Re-pack a KxN weight matrix into "paired-K" layout so a WMMA B-fragment
// (W[k][col], W[k+1][col]) is one aligned 8-byte load:
//   P[(k/2)*(2N) + 2*col + (k&1)] = W[k*N + col]
// ---------------------------------------------------------------------------
__global__ __launch_bounds__(256) void k_pair(
    const float* __restrict__ w, float* __restrict__ p, int K, int Ncols) {
  const int i = blockIdx.x * 256 + threadIdx.x;
  if (i < K * Ncols) {
    const int k = i / Ncols, col = i - k * Ncols;
    p[(k >> 1) * (2 * Ncols) + 2 * col + (k & 1)] = w[i];
  }
}

// ---------------------------------------------------------------------------
// h = ag = x @ in_w + in_b        [50000,176] @ [176,64]
// 8 waves/block; each wave owns a 16-row tile, 4 col-tiles of 16.
// in_w staged in LDS in paired-K layout (45 KB) -> B frag = 1x ds_load_b64.
// Dual-stores into h and ag (seeds the GINE self term, eps = 0).
// Also zeroes the 64-entry pooling accumulator.
// ---------------------------------------------------------------------------
__global__ __launch_bounds__(256) void k_in_proj(
    const float* __restrict__ x, const float* __restrict__ w,
    const float* __restrict__ b, float* __restrict__ h,
    float* __restrict__ ag, float* __restrict__ gpool) {
  __shared__ float sw[FF * HH];  // paired-K layout
  for (int i = threadIdx.x; i < FF * HH; i += 256) {
    const int k = i / HH, col = i - k * HH;
    sw[(k >> 1) * (2 * HH) + 2 * col + (k & 1)] = w[i];
  }
  if (blockIdx.x == 0 && threadIdx.x < HH) gpool[threadIdx.x] = 0.0f;
  __syncthreads();

  const int wave = threadIdx.x >> 5;
  const int lane = threadIdx.x & 31;
  const int half = lane >> 4;
  const int l16  = lane & 15;
  const int row0 = (blockIdx.x * 8 + wave) * 16;
  if (row0 >= NN) return;          // per-wave guard; rows within tile are valid
  const int rowa = row0 + l16;     // < NN always (NN multiple of 16)

  v8f acc[4] = {};
  for (int k0 = 0; k0 < FF; k0 += 4) {
    const int kk = k0 + 2 * half;            // even
    const v2f a = *(const v2f*)(x + (size_t)rowa * FF + kk);
#pragma unroll
    for (int c = 0; c < 4; ++c) {
      const int col = c * 16 + l16;
      const v2f bb = *(const v2f*)&sw[(kk >> 1) * (2 * HH) + 2 * col];
      acc[c] = wmma4(a, bb, acc[c]);
    }
  }
#pragma unroll
  for (int c = 0; c < 4; ++c) {
    const int col = c * 16 + l16;
    const float bias = b[col];
#pragma unroll
    for (int r = 0; r < 8; ++r) {
      const size_t gr = (size_t)(row0 + half * 8 + r);
      const float v = acc[c][r] + bias;
      h[gr * HH + col]  = v;
      ag[gr * HH + col] = v;
    }
  }
}

// ---------------------------------------------------------------------------
// Edge message + scatter:  aggr[dst] += relu(h[src] + edge_attr @ Wl + bl)
// One wave per edge; lane handles 2 channels (float2). Edge indices/attrs
// are wave-uniform -> readfirstlane forces them onto the scalar (SMEM) path,
// keeping the vector pipe free for the h-gather and the f32 atomics.
// ---------------------------------------------------------------------------
__global__ __launch_bounds__(256) void k_edge(
    const int* __restrict__ eidx, const float* __restrict__ eattr,
    const float* __restrict__ ew, const float* __restrict__ eb,
    const float* __restrict__ h, float* __restrict__ aggr) {
  __shared__ float sw[4 * HH];
  __shared__ float sb[HH];
  for (int i = threadIdx.x; i < 4 * HH; i += 256) sw[i] = ew[i];
  for (int i = threadIdx.x; i < HH; i += 256) sb[i] = eb[i];
  __syncthreads();

  const int lane = threadIdx.x & 31;
  const int c0 = lane * 2;
  const int wave = blockIdx.x * 8 + (threadIdx.x >> 5);
  const int nw = gridDim.x * 8;

  for (int e0 = wave; e0 < EE; e0 += nw) {
    const int e = __builtin_amdgcn_readfirstlane(e0);   // wave-uniform
    const int s = eidx[e];
    const int d = eidx[EE + e];
    const float4 ea = *(const float4*)(eattr + (size_t)e * 4);
    const float2 hs = *(const float2*)(h + (size_t)s * HH + c0);
    float m0 = sb[c0]     + hs.x + ea.x * sw[c0]       + ea.y * sw[HH + c0]
                          + ea.z * sw[2 * HH + c0]     + ea.w * sw[3 * HH + c0];
    float m1 = sb[c0 + 1] + hs.y + ea.x * sw[c0 + 1]   + ea.y * sw[HH + c0 + 1]
                          + ea.z * sw[2 * HH + c0 + 1] + ea.w * sw[3 * HH + c0 + 1];
    m0 = fmaxf(m0, 0.0f);
    m1 = fmaxf(m1, 0.0f);
    float* ap = aggr + (size_t)d * HH + c0;
    atom_fadd(ap, m0);
    atom_fadd(ap + 1, m1);
  }
}

// ---------------------------------------------------------------------------
// Node update (fused): z = aggr; t = relu(z@W1+b1); y = t@W2+b2;
// y = LN(y)*g + b; h = ag = relu(y). Last layer also pools into gpool.
// 4 waves/block; each wave owns a 16-row tile. 16x128 intermediate staged
// in padded LDS. Weights read as b64 fragments from paired global layout.
// ---------------------------------------------------------------------------
__global__ __launch_bounds__(128) void k_node(
    const float* __restrict__ z, const float* __restrict__ w1p,
    const float* __restrict__ b1, const float* __restrict__ w2p,
    const float* __restrict__ b2, const float* __restrict__ lng,
    const float* __restrict__ lnb, float* __restrict__ hout,
    float* __restrict__ agout, float* __restrict__ gpool, int isLast) {
  __shared__ float stg[4][16 * H2P];   // 33.8 KB

  const int wave = threadIdx.x >> 5;
  const int lane = threadIdx.x & 31;
  const int half = lane >> 4;
  const int l16  = lane & 15;
  const int row0 = (blockIdx.x * 4 + wave) * 16;
  if (row0 >= NN) return;          // per-wave guard; rows within tile are valid
  const int rowa = row0 + l16;     // < NN always

  // ---- phase 1: t = relu(z @ W1 + b1)   [16x64]@[64x128] ----
  {
    v8f a1[8] = {};
    for (int k0 = 0; k0 < HH; k0 += 4) {
      const int kk = k0 + 2 * half;     // even
      const v2f a = *(const v2f*)(z + (size_t)rowa * HH + kk);
      const float* wrow = w1p + (size_t)(kk >> 1) * (2 * H2);
#pragma unroll
      for (int c = 0; c < 8; ++c) {
        const v2f bb = *(const v2f*)(wrow + 2 * (c * 16 + l16));
        a1[c] = wmma4(a, bb, a1[c]);
      }
    }
#pragma unroll
    for (int c = 0; c < 8; ++c) {
      const int col = c * 16 + l16;
      const float bias = b1[col];
#pragma unroll
      for (int r = 0; r < 8; ++r)
        stg[wave][(half * 8 + r) * H2P + col] = fmaxf(a1[c][r] + bias, 0.0f);
    }
  }
  // same-wave LDS RAW: DS ops are in-order within a wave; no barrier needed.

  // ---- phase 2: y = t @ W2 + b2   [16x128]@[128x64] ----
  v8f a2[4] = {};
  for (int k0 = 0; k0 < H2; k0 += 4) {
    const int kk = k0 + 2 * half;       // even
    const v2f a = *(const v2f*)&stg[wave][l16 * H2P + kk];
    const float* wrow = w2p + (size_t)(kk >> 1) * (2 * HH);
#pragma unroll
    for (int c = 0; c < 4; ++c) {
      const v2f bb = *(const v2f*)(wrow + 2 * (c * 16 + l16));
      a2[c] = wmma4(a, bb, a2[c]);
    }
  }

  // ---- LayerNorm + relu (+ pooling on last layer) ----
  float psum[4] = {0.0f, 0.0f, 0.0f, 0.0f};
#pragma unroll
  for (int r = 0; r < 8; ++r) {
    float vs[4], s = 0.0f, ss = 0.0f;
#pragma unroll
    for (int c = 0; c < 4; ++c) {
      const int col = c * 16 + l16;
      const float v = a2[c][r] + b2[col];
      vs[c] = v; s += v; ss += v * v;
    }
#pragma unroll
    for (int m = 1; m < 16; m <<= 1) {   // reduce across the 16-lane half
      s  += __shfl_xor(s, m, 32);
      ss += __shfl_xor(ss, m, 32);
    }
    const float mu  = s * (1.0f / HH);
    const float var = ss * (1.0f / HH) - mu * mu;
    const float inv = rsqrtf(var + 1e-5f);
    const size_t gr = (size_t)(row0 + half * 8 + r);
#pragma unroll
    for (int c = 0; c < 4; ++c) {
      const int col = c * 16 + l16;
      const float y = fmaxf((vs[c] - mu) * inv * lng[col] + lnb[col], 0.0f);
      hout[gr * HH + col]  = y;   // next layer's h
      agout[gr * HH + col] = y;   // seed next layer's aggregate
      psum[c] += y;
    }
  }
  if (isLast) {
#pragma unroll
    for (int c = 0; c < 4; ++c) atom_fadd(&gpool[c * 16 + l16], psum[c]);
  }
}

// ---------------------------------------------------------------------------
// Head: out = relu(g @ reg1 + b1) @ reg2 + b2   (scalar)
// ---------------------------------------------------------------------------
__global__ __launch_bounds__(64) void k_head(
    const float* __restrict__ gpool, const float* __restrict__ w1,
    const float* __restrict__ b1, const float* __restrict__ w2,
    const float* __restrict__ b2, float* __restrict__ out) {
  __shared__ float sg[HH];
  __shared__ float sh[HH];
  const int t = threadIdx.x;
  sg[t] = gpool[t];
  __syncthreads();
  float acc = b1[t];
  for (int c = 0; c < HH; ++c) acc += sg[c] * w1[c * HH + t];
  sh[t] = fmaxf(acc, 0.0f) * w2[t];
  __syncthreads();
  if (t == 0) {
    float s = b2[0];
    for (int j = 0; j < HH; ++j) s += sh[j];
    out[0] = s;
  }
}

// ---------------------------------------------------------------------------
extern "C" void kernel_launch(void* const* d_in, const int* in_sizes, int n_in,
                              void* d_out, int out_size, void* d_ws, size_t ws_size,
                              hipStream_t stream) {
  const float* x      = (const float*)d_in[0];
  const int*   eidx   = (const int*)  d_in[1];
  const float* eattr  = (const float*)d_in[2];
  const float* in_w   = (const float*)d_in[3];
  const float* in_b   = (const float*)d_in[4];
  const float* edge_w = (const float*)d_in[5];
  const float* edge_b = (const float*)d_in[6];
  const float* mlp1_w = (const float*)d_in[7];
  const float* mlp1_b = (const float*)d_in[8];
  const float* mlp2_w = (const float*)d_in[9];
  const float* mlp2_b = (const float*)d_in[10];
  const float* ln_g   = (const float*)d_in[11];
  const float* ln_b   = (const float*)d_in[12];
  const float* reg1_w = (const float*)d_in[13];
  const float* reg1_b = (const float*)d_in[14];
  const float* reg2_w = (const float*)d_in[15];
  const float* reg2_b = (const float*)d_in[16];
  (void)in_sizes; (void)n_in; (void)out_size; (void)ws_size;

  float* ws  = (float*)d_ws;
  float* h   = ws;                                   // [N,H]  12.8 MB
  float* ag  = ws + (size_t)NN * HH;                 // [N,H]  12.8 MB
  float* gp  = ws + (size_t)2 * NN * HH;             // [64]
  float* w1p = gp + HH;                              // [L,64,128] paired
  float* w2p = w1p + (size_t)4 * HH * H2;            // [L,128,64] paired

  const int WE = HH * H2;                            // 8192 elems per matrix
  for (int l = 0; l < 4; ++l) {
    k_pair<<<(WE + 255) / 256, 256, 0, stream>>>(mlp1_w + (size_t)l * WE,
                                                 w1p + (size_t)l * WE, HH, H2);
    k_pair<<<(WE + 255) / 256, 256, 0, stream>>>(mlp2_w + (size_t)l * WE,
                                                 w2p + (size_t)l * WE, H2, HH);
  }

  const int nTiles = (NN + 15) / 16;                 // 3125
  k_in_proj<<<(nTiles + 7) / 8, 256, 0, stream>>>(x, in_w, in_b, h, ag, gp);

  for (int l = 0; l < 4; ++l) {
    k_edge<<<4096, 256, 0, stream>>>(eidx, eattr,
                                     edge_w + (size_t)l * 4 * HH,
                                     edge_b + (size_t)l * HH, h, ag);
    k_node<<<(nTiles + 3) / 4, 128, 0, stream>>>(
        ag, w1p + (size_t)l * WE, mlp1_b + (size_t)l * H2,
        w2p + (size_t)l * WE, mlp2_b + (size_t)l * HH,
        ln_g + (size_t)l * HH, ln_b + (size_t)l * HH, h, ag, gp,
        l == 3 ? 1 : 0);
  }
  k_head<<<1, 64, 0, stream>>>(gp, reg1_w, reg1_b, reg2_w, reg2_b, (float*)d_out);
}